// GNNVAE_58712202936396
// MI455X (gfx1250) — compile-verified
//
#include <hip/hip_runtime.h>

typedef __attribute__((ext_vector_type(2))) float v2f;
typedef __attribute__((ext_vector_type(8))) float v8f;

#define NN 50000
#define NE 800000
#define KIN 128
#define HID 64
#define NC 32
#define ROW_TILES (NN / 16)   // 3125, exact

// ---------------- degree / normalization ----------------
__global__ void k_zero(float* __restrict__ p, int n) {
  int i = blockIdx.x * blockDim.x + threadIdx.x;
  if (i < n) p[i] = 0.0f;
}

__global__ void k_degree(const int* __restrict__ dst, float* __restrict__ deg) {
  int e = blockIdx.x * blockDim.x + threadIdx.x;
  if (e < NE) atomicAdd(&deg[dst[e]], 1.0f);
}

__global__ void k_dinv(float* __restrict__ deg) {
  int i = blockIdx.x * blockDim.x + threadIdx.x;
  if (i < NN) deg[i] = rsqrtf(deg[i] + 1.0f);   // deg >= 1, finite
}

// ---------------- layer-1 GEMM: H = X @ W0  (fp32 WMMA 16x16x4) ----------------
// X: [NN, 128] row-major, W0: [128, 64] row-major, H: [NN, 64]
__global__ void k_gemm_xw0(const float* __restrict__ X,
                           const float* __restrict__ W,
                           float* __restrict__ H) {
  int wave = (blockIdx.x * blockDim.x + threadIdx.x) >> 5;
  if (wave >= ROW_TILES) return;          // wave-uniform: EXEC stays all-ones
  int lane = threadIdx.x & 31;
  int hh   = lane >> 4;                   // 0: K={0,1}, 1: K={2,3}
  int lid  = lane & 15;
  const float* xr = X + (wave * 16 + lid) * KIN;  // A row = lid
  v8f c0 = {}, c1 = {}, c2 = {}, c3 = {};
  for (int k = 0; k < KIN; k += 4) {
    __builtin_prefetch(xr + k + 32, 0, 1);        // global_prefetch_b8
    int ka = k + 2 * hh;
    v2f a;
    a.x = xr[ka];
    a.y = xr[ka + 1];
    const float* wr0 = W + ka * HID;              // W row ka
    const float* wr1 = wr0 + HID;                 // W row ka+1
    v2f b;
    b.x = wr0[lid];      b.y = wr1[lid];
    c0 = __builtin_amdgcn_wmma_f32_16x16x4_f32(false, a, false, b, (short)0, c0, false, false);
    b.x = wr0[lid + 16]; b.y = wr1[lid + 16];
    c1 = __builtin_amdgcn_wmma_f32_16x16x4_f32(false, a, false, b, (short)0, c1, false, false);
    b.x = wr0[lid + 32]; b.y = wr1[lid + 32];
    c2 = __builtin_amdgcn_wmma_f32_16x16x4_f32(false, a, false, b, (short)0, c2, false, false);
    b.x = wr0[lid + 48]; b.y = wr1[lid + 48];
    c3 = __builtin_amdgcn_wmma_f32_16x16x4_f32(false, a, false, b, (short)0, c3, false, false);
  }
  // D layout: VGPR j -> row (hh?8+j:j), col = lid (lanes 0-15) / lid (lanes 16-31)
  int mrow = wave * 16 + hh * 8;
  for (int j = 0; j < 8; ++j) {
    float* o = H + (mrow + j) * HID + lid;
    o[0]  = c0[j];
    o[16] = c1[j];
    o[32] = c2[j];
    o[48] = c3[j];
  }
}

// ---------------- layers 2&3 fused GEMM: MU = H@W1, LV = H@W2 ----------------
// H: [NN, 64], W1/W2: [64, 32], outputs [NN, 32]
__global__ void k_gemm_hw12(const float* __restrict__ H,
                            const float* __restrict__ W1,
                            const float* __restrict__ W2,
                            float* __restrict__ MU,
                            float* __restrict__ LV) {
  int wave = (blockIdx.x * blockDim.x + threadIdx.x) >> 5;
  if (wave >= ROW_TILES) return;
  int lane = threadIdx.x & 31;
  int hh   = lane >> 4;
  int lid  = lane & 15;
  const float* hr = H + (wave * 16 + lid) * HID;
  v8f m0 = {}, m1 = {}, l0 = {}, l1 = {};
  for (int k = 0; k < HID; k += 4) {
    int ka = k + 2 * hh;
    v2f a;
    a.x = hr[ka];
    a.y = hr[ka + 1];
    const float* u0 = W1 + ka * NC;
    const float* u1 = u0 + NC;
    const float* v0 = W2 + ka * NC;
    const float* v1 = v0 + NC;
    v2f b;
    b.x = u0[lid];      b.y = u1[lid];
    m0 = __builtin_amdgcn_wmma_f32_16x16x4_f32(false, a, false, b, (short)0, m0, false, false);
    b.x = u0[lid + 16]; b.y = u1[lid + 16];
    m1 = __builtin_amdgcn_wmma_f32_16x16x4_f32(false, a, false, b, (short)0, m1, false, false);
    b.x = v0[lid];      b.y = v1[lid];
    l0 = __builtin_amdgcn_wmma_f32_16x16x4_f32(false, a, false, b, (short)0, l0, false, false);
    b.x = v0[lid + 16]; b.y = v1[lid + 16];
    l1 = __builtin_amdgcn_wmma_f32_16x16x4_f32(false, a, false, b, (short)0, l1, false, false);
  }
  int mrow = wave * 16 + hh * 8;
  for (int j = 0; j < 8; ++j) {
    float* om = MU + (mrow + j) * NC + lid;
    float* ol = LV + (mrow + j) * NC + lid;
    om[0]  = m0[j];
    om[16] = m1[j];
    ol[0]  = l0[j];
    ol[16] = l1[j];
  }
}

// ---------------- aggregation: self-loop init + edge scatter ----------------
__global__ void k_selfinit64(const float* __restrict__ Hlin,
                             const float* __restrict__ dinv,
                             const float* __restrict__ bias,
                             float* __restrict__ Agg) {
  int idx = blockIdx.x * blockDim.x + threadIdx.x;
  if (idx >= NN * HID) return;
  int i = idx >> 6;
  int f = idx & 63;
  float d = dinv[i];
  Agg[idx] = Hlin[idx] * d * d + bias[f];
}

__global__ void k_scatter64(const float* __restrict__ Hlin,
                            const int* __restrict__ src,
                            const int* __restrict__ dst,
                            const float* __restrict__ dinv,
                            float* __restrict__ Agg) {
  int idx = blockIdx.x * blockDim.x + threadIdx.x;
  int e = idx >> 6;
  if (e >= NE) return;
  int f = idx & 63;
  int s = src[e], d = dst[e];
  float nrm = dinv[s] * dinv[d];
  atomicAdd(&Agg[d * HID + f], Hlin[s * HID + f] * nrm);  // global_atomic_add_f32
}

__global__ void k_reludrop(const float* __restrict__ Agg,
                           const float* __restrict__ mask,
                           float* __restrict__ Hout) {
  int idx = blockIdx.x * blockDim.x + threadIdx.x;
  if (idx < NN * HID) Hout[idx] = fmaxf(Agg[idx], 0.0f) * mask[idx];
}

__global__ void k_selfinit32(const float* __restrict__ MUlin,
                             const float* __restrict__ LVlin,
                             const float* __restrict__ dinv,
                             const float* __restrict__ b1,
                             const float* __restrict__ b2,
                             float* __restrict__ MUout,
                             float* __restrict__ LVout) {
  int idx = blockIdx.x * blockDim.x + threadIdx.x;
  if (idx >= NN * NC) return;
  int i = idx >> 5;
  int f = idx & 31;
  float d  = dinv[i];
  float dd = d * d;
  MUout[idx] = MUlin[idx] * dd + b1[f];
  LVout[idx] = LVlin[idx] * dd + b2[f];
}

__global__ void k_scatter32(const float* __restrict__ MUlin,
                            const float* __restrict__ LVlin,
                            const int* __restrict__ src,
                            const int* __restrict__ dst,
                            const float* __restrict__ dinv,
                            float* __restrict__ MUout,
                            float* __restrict__ LVout) {
  int idx = blockIdx.x * blockDim.x + threadIdx.x;
  int e = idx >> 5;
  if (e >= NE) return;
  int f = idx & 31;
  int s = src[e], d = dst[e];
  float nrm = dinv[s] * dinv[d];
  atomicAdd(&MUout[d * NC + f], MUlin[s * NC + f] * nrm);
  atomicAdd(&LVout[d * NC + f], LVlin[s * NC + f] * nrm);
}

__global__ void k_final(const float* __restrict__ eps,
                        const float* __restrict__ MU,
                        const float* __restrict__ LV,
                        float* __restrict__ Z) {
  int idx = blockIdx.x * blockDim.x + threadIdx.x;
  if (idx < NN * NC) Z[idx] = eps[idx] * expf(LV[idx]) + MU[idx];
}

// ---------------- driver ----------------
extern "C" void kernel_launch(void* const* d_in, const int* in_sizes, int n_in,
                              void* d_out, int out_size, void* d_ws, size_t ws_size,
                              hipStream_t stream) {
  (void)in_sizes; (void)n_in; (void)out_size; (void)ws_size;
  const float* x    = (const float*)d_in[0];
  const int*   ei   = (const int*)d_in[1];
  const float* W0   = (const float*)d_in[2];
  const float* b0   = (const float*)d_in[3];
  const float* W1   = (const float*)d_in[4];
  const float* b1   = (const float*)d_in[5];
  const float* W2   = (const float*)d_in[6];
  const float* b2   = (const float*)d_in[7];
  const float* mask = (const float*)d_in[8];
  const float* eps  = (const float*)d_in[9];
  const int* src = ei;          // edge_index[0]
  const int* dst = ei + NE;     // edge_index[1]

  float* ws    = (float*)d_ws;
  float* dinv  = ws;                    // NN
  float* hlin  = dinv + NN;             // NN*HID
  float* hagg  = hlin + (size_t)NN * HID;   // NN*HID
  float* mulin = hagg + (size_t)NN * HID;   // NN*NC
  float* lvlin = mulin + (size_t)NN * NC;   // NN*NC

  float* z  = (float*)d_out;            // [NN, NC]
  float* mu = z + (size_t)NN * NC;      // [NN, NC]
  float* lv = mu + (size_t)NN * NC;     // [NN, NC]

  const int B = 256;
  const int gN   = (NN + B - 1) / B;
  const int gE   = (NE + B - 1) / B;
  const int gW   = (ROW_TILES * 32 + B - 1) / B;      // 8 waves / block
  const int gNH  = (NN * HID + B - 1) / B;
  const int gEH  = (NE * HID + B - 1) / B;            // wraps int? 51.2M < 2^31, ok
  const int gNC_ = (NN * NC + B - 1) / B;
  const int gEC  = (NE * NC + B - 1) / B;

  // degree -> dinv
  k_zero   <<<gN, B, 0, stream>>>(dinv, NN);
  k_degree <<<gE, B, 0, stream>>>(dst, dinv);
  k_dinv   <<<gN, B, 0, stream>>>(dinv);

  // layer 1
  k_gemm_xw0  <<<gW,  B, 0, stream>>>(x, W0, hlin);
  k_selfinit64<<<gNH, B, 0, stream>>>(hlin, dinv, b0, hagg);
  k_scatter64 <<<gEH, B, 0, stream>>>(hlin, src, dst, dinv, hagg);
  k_reludrop  <<<gNH, B, 0, stream>>>(hagg, mask, hlin);   // hlin reused as h

  // layers 2 & 3 (fused)
  k_gemm_hw12 <<<gW,   B, 0, stream>>>(hlin, W1, W2, mulin, lvlin);
  k_selfinit32<<<gNC_, B, 0, stream>>>(mulin, lvlin, dinv, b1, b2, mu, lv);
  k_scatter32 <<<gEC,  B, 0, stream>>>(mulin, lvlin, src, dst, dinv, mu, lv);

  // reparameterize
  k_final     <<<gNC_, B, 0, stream>>>(eps, mu, lv, z);
}